// PCDConv_62362925138477
// MI455X (gfx1250) — compile-verified
//
#include <hip/hip_runtime.h>
#include <hip/hip_bf16.h>

// Problem constants (match reference)
#define BB     8
#define NN     4096
#define CIN    64
#define COUT   64
#define KNN    16
#define DREAL  67          // 3 + 64
#define DP     68          // padded to multiple of 4 (17 WMMA K-steps of 4)
#define KSTEPS 17          // DP / 4
#define FLTMAX 3.402823466e+38f

typedef __attribute__((ext_vector_type(2))) float v2f;
typedef __attribute__((ext_vector_type(8))) float v8f;

// ---------------------------------------------------------------------------
// Kernel 1: build padded node-major feature rows f[b][n][0..67] and sq norms.
// ---------------------------------------------------------------------------
__global__ void build_f_kernel(const float* __restrict__ x_loc,
                               const float* __restrict__ x_feat,
                               float* __restrict__ fws,
                               float* __restrict__ sqws) {
    int gid = blockIdx.x * blockDim.x + threadIdx.x;
    if (gid >= BB * NN) return;
    int b = gid >> 12;          // / 4096
    int n = gid & (NN - 1);
    float* row = fws + (size_t)gid * DP;
    float s = 0.0f;
#pragma unroll
    for (int d = 0; d < 3; ++d) {
        float v = x_loc[((size_t)b * 3 + d) * NN + n];
        row[d] = v; s += v * v;
    }
#pragma unroll 8
    for (int c = 0; c < CIN; ++c) {
        float v = x_feat[((size_t)b * CIN + c) * NN + n];
        row[3 + c] = v; s += v * v;
    }
    row[DREAL] = 0.0f;          // pad
    sqws[gid] = s;
}

// ---------------------------------------------------------------------------
// Register-resident top-k=16 insertion (fully unrolled -> stays in VGPRs).
// ---------------------------------------------------------------------------
__device__ __forceinline__ void topk_insert(float (&td)[KNN], int (&ti)[KNN],
                                            float& tmax, float d, int idx) {
    if (d < tmax) {
        // find slot holding current max
        int am = 0; float mv = td[0];
#pragma unroll
        for (int i = 1; i < KNN; ++i) { if (td[i] > mv) { mv = td[i]; am = i; } }
#pragma unroll
        for (int i = 0; i < KNN; ++i) { if (i == am) { td[i] = d; ti[i] = idx; } }
        float nm = td[0];
#pragma unroll
        for (int i = 1; i < KNN; ++i) nm = fmaxf(nm, td[i]);
        tmax = nm;
    }
}

// ---------------------------------------------------------------------------
// Kernel 2: fused distance-GEMM (WMMA f32 16x16x4) + top-16 + gather-sum +
//           linear (W_rel, W_root) + bias + ReLU.
// Grid: (N/16, B). Block: 128 threads = 4 waves. Wave w owns column tiles
// w, w+4, ... ; all waves share the same 16-row tile.
// B-fragments are staged into a register array so the 17 global_load_b64s
// issue as one clause (single wait), then 17 WMMAs run back-to-back.
// ---------------------------------------------------------------------------
__global__ void __launch_bounds__(128)
knn_graphconv_kernel(const float* __restrict__ fws,
                     const float* __restrict__ sqws,
                     const float* __restrict__ W_rel,
                     const float* __restrict__ W_root,
                     const float* __restrict__ bias,
                     float* __restrict__ outF) {
    __shared__ float dtile[4][16][17];     // per-wave distance tile (pad 17)
    __shared__ float cand_d[16][128];      // 8 partial lists x 16 per row
    __shared__ int   cand_i[16][128];
    __shared__ int   fidx[16][KNN];        // final neighbor indices
    __shared__ float neigh[16][DP];        // summed neighbor features

    const int tid      = threadIdx.x;
    const int lane     = tid & 31;
    const int w        = tid >> 5;         // wave id 0..3
    const int lm       = lane & 15;        // lane % 16
    const int hi       = lane >> 4;        // 0 or 1
    const int b        = blockIdx.y;
    const int row_base = blockIdx.x * 16;

    const float* fb  = fws  + (size_t)b * NN * DP;
    const float* sqb = sqws + (size_t)b * NN;

    // ---- preload A fragments (rows row_base..row_base+15), 17 K-steps ----
    // ISA layout: lanes 0-15 hold M=lane, K={0,1}; lanes 16-31 hold K={2,3}.
    const float* arow = fb + (size_t)(row_base + lm) * DP + 2 * hi;
    v2f afrag[KSTEPS];
#pragma unroll
    for (int kk = 0; kk < KSTEPS; ++kk)
        afrag[kk] = *(const v2f*)(arow + 4 * kk);

    // squared norms of the 8 rows this lane's C-fragment covers
    float rowsq[8];
#pragma unroll
    for (int v = 0; v < 8; ++v)
        rowsq[v] = sqb[row_base + 8 * hi + v];

    // per-lane top-16 partial list (lane scans half a row per tile)
    float td[KNN]; int ti[KNN]; float tmax = FLTMAX;
#pragma unroll
    for (int i = 0; i < KNN; ++i) { td[i] = FLTMAX; ti[i] = 0; }

    const int myrow = lm;          // row this lane scans
    const int jb    = 8 * hi;      // column half this lane scans

    // ---- column-tile loop ----
    for (int t = w; t < NN / 16; t += 4) {
        const int col_base = t * 16;

        // stage all 17 B fragments first: one load clause, one wait
        const float* brow = fb + (size_t)(col_base + lm) * DP + 2 * hi;
        v2f bfrag[KSTEPS];
#pragma unroll
        for (int kk = 0; kk < KSTEPS; ++kk)
            bfrag[kk] = *(const v2f*)(brow + 4 * kk);
        const int   gcol = col_base + lm;
        const float sqc  = sqb[gcol];

        // accumulate 16x16 dot-product tile: 17 back-to-back f32 WMMAs
        v8f c = {};
#pragma unroll
        for (int kk = 0; kk < KSTEPS; ++kk)
            c = __builtin_amdgcn_wmma_f32_16x16x4_f32(
                    false, afrag[kk], false, bfrag[kk], (short)0, c,
                    false, false);

        // distances -> LDS (lane holds column j=lm, rows 8*hi..8*hi+7)
#pragma unroll
        for (int v = 0; v < 8; ++v) {
            int   grow = row_base + 8 * hi + v;
            float dv   = rowsq[v] + sqc - 2.0f * c[v];
            if (grow == gcol) dv = FLTMAX;          // exclude self
            dtile[w][8 * hi + v][lm] = dv;
        }
        // same-wave LDS ops are in-order: safe to read back without barrier
#pragma unroll
        for (int jj = 0; jj < 8; ++jj) {
            float dv = dtile[w][myrow][jb + jj];
            topk_insert(td, ti, tmax, dv, col_base + jb + jj);
        }
    }

    // ---- publish partial lists: list id = 2*w + hi, row = lm ----
    {
        int base = (2 * w + hi) * KNN;
#pragma unroll
        for (int i = 0; i < KNN; ++i) {
            cand_d[lm][base + i] = td[i];
            cand_i[lm][base + i] = ti[i];
        }
    }
    __syncthreads();

    // ---- merge 8 partial lists per row (threads 0..15, one row each) ----
    if (tid < 16) {
        float md[KNN]; int mi[KNN]; float mmax = FLTMAX;
#pragma unroll
        for (int i = 0; i < KNN; ++i) { md[i] = FLTMAX; mi[i] = 0; }
        for (int p = 0; p < 128; ++p)
            topk_insert(md, mi, mmax, cand_d[tid][p], cand_i[tid][p]);
#pragma unroll
        for (int i = 0; i < KNN; ++i) fidx[tid][i] = mi[i];
    }
    __syncthreads();

    // ---- neighbor gather + sum: 16 rows x 68 dims ----
    for (int task = tid; task < 16 * DP; task += 128) {
        int r = task / DP, d = task % DP;
        float s = 0.0f;
#pragma unroll
        for (int k = 0; k < KNN; ++k)
            s += fb[(size_t)fidx[r][k] * DP + d];
        neigh[r][d] = s;
    }
    __syncthreads();

    // ---- linear + bias + ReLU: 16 rows x 64 out channels ----
    for (int task = tid; task < 16 * COUT; task += 128) {
        int r = task >> 6, c = task & 63;
        const float* frow = fb + (size_t)(row_base + r) * DP;
        const float* wr = W_rel  + c * DREAL;
        const float* wt = W_root + c * DREAL;
        float acc = bias[c];
#pragma unroll 1
        for (int d = 0; d < DREAL; ++d)
            acc = fmaf(wr[d], neigh[r][d], fmaf(wt[d], frow[d], acc));
        acc = fmaxf(acc, 0.0f);
        outF[((size_t)b * COUT + c) * NN + (row_base + r)] = acc;
    }
}

// ---------------------------------------------------------------------------
extern "C" void kernel_launch(void* const* d_in, const int* in_sizes, int n_in,
                              void* d_out, int out_size, void* d_ws, size_t ws_size,
                              hipStream_t stream) {
    const float* x_loc  = (const float*)d_in[0];   // [B,3,N]
    const float* x_feat = (const float*)d_in[1];   // [B,64,N]
    const float* W_rel  = (const float*)d_in[2];   // [64,67]
    const float* W_root = (const float*)d_in[3];   // [64,67]
    const float* bias   = (const float*)d_in[4];   // [64]
    (void)in_sizes; (void)n_in; (void)out_size;

    float* fws  = (float*)d_ws;                       // B*N*DP floats
    float* sqws = fws + (size_t)BB * NN * DP;         // B*N floats
    (void)ws_size;

    float* out  = (float*)d_out;
    // output tuple: (x_loc passthrough, conv output)
    hipMemcpyAsync(out, x_loc, (size_t)BB * 3 * NN * sizeof(float),
                   hipMemcpyDeviceToDevice, stream);
    float* outF = out + (size_t)BB * 3 * NN;

    build_f_kernel<<<(BB * NN + 255) / 256, 256, 0, stream>>>(
        x_loc, x_feat, fws, sqws);

    dim3 grid(NN / 16, BB);
    knn_graphconv_kernel<<<grid, 128, 0, stream>>>(
        fws, sqws, W_rel, W_root, bias, outF);
}